// Trans_GATV2_60198261621557
// MI455X (gfx1250) — compile-verified
//
#include <hip/hip_runtime.h>
#include <hip/hip_bf16.h>
#include <math.h>

typedef __attribute__((ext_vector_type(2))) float v2f;
typedef __attribute__((ext_vector_type(8))) float v8f;

#define NEG_SLOPE 0.2f

__device__ __forceinline__ float warpSum(float v) {
#pragma unroll
  for (int off = 16; off; off >>= 1) v += __shfl_xor(v, off, 32);
  return v;
}

// float atomic-max via ordered-int trick (buffer must be initialized to -inf)
__device__ __forceinline__ void atomicMaxF(float* addr, float val) {
  if (val >= 0.0f) atomicMax((int*)addr, __float_as_int(val));
  else             atomicMin((unsigned int*)addr, __float_as_uint(val));
}

__global__ void fill_kernel(float* __restrict__ p, long long n, float v) {
  long long i = (long long)blockIdx.x * blockDim.x + threadIdx.x;
  if (i < n) p[i] = v;
}

// ---------------- WMMA f32 GEMM: C[M,Nout] = A[M,K] @ B[K,Nout] ----------------
// One wave computes a 32 x (NT*16) output tile (2 M-blocks x NT N-tiles) with
// v_wmma_f32_16x16x4_f32, K stepped by 4 and register double-buffered so the
// next step's A/B loads are in flight while the current step's WMMAs issue.
// Requires M % 16 == 0, K % 4 == 0 (and K >= 4), Nout % 16 == 0.
template<int NT>
__global__ void gemm_wmma_f32(const float* __restrict__ A, const float* __restrict__ B,
                              float* __restrict__ C, int M, int K, int Nout) {
  int wave = (int)(((long long)blockIdx.x * blockDim.x + threadIdx.x) >> 5);
  int lane = threadIdx.x & 31;
  int mTiles = (M + 31) >> 5;            // 32-row tiles; last may hold only 16 rows
  int mT = wave % mTiles;
  int nG = wave / mTiles;
  int n0 = nG * (NT * 16);
  if (n0 >= Nout) return;                // wave-uniform guard (EXEC stays all-1)
  int r  = lane & 15;                    // A-row / B-col / C-col lane index
  int kq = (lane >> 4) << 1;             // K sub-offset from lane half
  int m0 = mT * 32;
  bool two = (m0 + 16) < M;              // second 16-row block valid (wave-uniform)

  const float* Ar0 = A + (size_t)(m0 + r) * K + kq;
  const float* Ar1 = Ar0 + (two ? (size_t)16 * K : 0);  // alias block0 when absent
  const float* Bb  = B + (size_t)kq * Nout + n0 + r;

  v8f acc0[NT], acc1[NT];
#pragma unroll
  for (int t = 0; t < NT; ++t) {
    acc0[t] = (v8f){0.f,0.f,0.f,0.f,0.f,0.f,0.f,0.f};
    acc1[t] = (v8f){0.f,0.f,0.f,0.f,0.f,0.f,0.f,0.f};
  }

  // prime kb = 0
  v2f a0, a1, bc[NT];
  a0.x = Ar0[0]; a0.y = Ar0[1];
  a1.x = Ar1[0]; a1.y = Ar1[1];
#pragma unroll
  for (int t = 0; t < NT; ++t) { bc[t].x = Bb[t * 16]; bc[t].y = Bb[t * 16 + Nout]; }

  for (int kb = 0; kb < K - 4; kb += 4) {
    int nk = kb + 4;
    // issue next step's loads before consuming current registers
    v2f na0, na1, nb[NT];
    na0.x = Ar0[nk]; na0.y = Ar0[nk + 1];
    na1.x = Ar1[nk]; na1.y = Ar1[nk + 1];
    const float* Bp = Bb + (size_t)nk * Nout;
#pragma unroll
    for (int t = 0; t < NT; ++t) { nb[t].x = Bp[t * 16]; nb[t].y = Bp[t * 16 + Nout]; }

#pragma unroll
    for (int t = 0; t < NT; ++t) {
      acc0[t] = __builtin_amdgcn_wmma_f32_16x16x4_f32(
          false, a0, false, bc[t], (short)0, acc0[t], false, false);
      acc1[t] = __builtin_amdgcn_wmma_f32_16x16x4_f32(
          false, a1, false, bc[t], (short)0, acc1[t], false, false);
    }
    a0 = na0; a1 = na1;
#pragma unroll
    for (int t = 0; t < NT; ++t) bc[t] = nb[t];
  }
  // drain last step
#pragma unroll
  for (int t = 0; t < NT; ++t) {
    acc0[t] = __builtin_amdgcn_wmma_f32_16x16x4_f32(
        false, a0, false, bc[t], (short)0, acc0[t], false, false);
    acc1[t] = __builtin_amdgcn_wmma_f32_16x16x4_f32(
        false, a1, false, bc[t], (short)0, acc1[t], false, false);
  }

  int mrow = m0 + ((lane >> 4) << 3);    // C: M = v + 8*(lane>=16)
#pragma unroll
  for (int t = 0; t < NT; ++t)
#pragma unroll
    for (int v = 0; v < 8; ++v)
      C[(size_t)(mrow + v) * Nout + n0 + t * 16 + r] = acc0[t][v];
  if (two) {
#pragma unroll
    for (int t = 0; t < NT; ++t)
#pragma unroll
      for (int v = 0; v < 8; ++v)
        C[(size_t)(mrow + 16 + v) * Nout + n0 + t * 16 + r] = acc1[t][v];
  }
}

// ---------------- Layer-1 edge kernels (H=8, Fo=32, F=256) ----------------
__global__ void attn1_kernel(const float* __restrict__ xl, const float* __restrict__ xr,
                             const int* __restrict__ srcI, const int* __restrict__ dstI,
                             const float* __restrict__ att,
                             float* __restrict__ ebuf, float* __restrict__ mbuf,
                             long long Ereal, long long Etot) {
  long long wv = (((long long)blockIdx.x * blockDim.x + threadIdx.x) >> 5);
  int lane = threadIdx.x & 31;
  if (wv >= Etot) return;
  int s, d;
  if (wv < Ereal) { s = srcI[wv]; d = dstI[wv]; } else { s = d = (int)(wv - Ereal); }
  const float* xs = xl + (size_t)s * 256;
  const float* xd = xr + (size_t)d * 256;
#pragma unroll
  for (int h = 0; h < 8; ++h) {
    int f = h * 32 + lane;
    float v = xs[f] + xd[f];
    v = v > 0.f ? v : NEG_SLOPE * v;
    v *= att[f];
    float sm = warpSum(v);
    if (lane == 0) {
      ebuf[wv * 8 + h] = sm;
      atomicMaxF(&mbuf[(size_t)d * 8 + h], sm);
    }
  }
}

__global__ void softden1_kernel(float* __restrict__ ebuf, float* __restrict__ den,
                                const int* __restrict__ dstI, const float* __restrict__ mbuf,
                                long long Ereal, long long Etot) {
  long long e = (long long)blockIdx.x * blockDim.x + threadIdx.x;
  if (e >= Etot) return;
  int d = (e < Ereal) ? dstI[e] : (int)(e - Ereal);
#pragma unroll
  for (int h = 0; h < 8; ++h) {
    float ex = expf(ebuf[e * 8 + h] - mbuf[(size_t)d * 8 + h]);
    ebuf[e * 8 + h] = ex;
    atomicAdd(&den[(size_t)d * 8 + h], ex);
  }
}

__global__ void agg1_kernel(const float* __restrict__ xl, const float* __restrict__ ebuf,
                            const float* __restrict__ den,
                            const int* __restrict__ srcI, const int* __restrict__ dstI,
                            float* __restrict__ outp, long long Ereal, long long Etot) {
  long long wv = (((long long)blockIdx.x * blockDim.x + threadIdx.x) >> 5);
  int lane = threadIdx.x & 31;
  if (wv >= Etot) return;
  int s, d;
  if (wv < Ereal) { s = srcI[wv]; d = dstI[wv]; } else { s = d = (int)(wv - Ereal); }
  const float* xs = xl + (size_t)s * 256;
  float* od = outp + (size_t)d * 256;
#pragma unroll
  for (int h = 0; h < 8; ++h) {
    float alpha = ebuf[wv * 8 + h] / (den[(size_t)d * 8 + h] + 1e-16f);
    int f = h * 32 + lane;
    atomicAdd(&od[f], xs[f] * alpha);
  }
}

__global__ void bias_elu_kernel(float* __restrict__ hb, const float* __restrict__ bias,
                                long long n, int F) {
  long long i = (long long)blockIdx.x * blockDim.x + threadIdx.x;
  if (i >= n) return;
  float v = hb[i] + bias[i % F];
  hb[i] = v > 0.f ? v : (expf(v) - 1.0f);
}

__global__ void bias_kernel(const float* __restrict__ in, float* __restrict__ outp,
                            const float* __restrict__ bias, long long n, int F) {
  long long i = (long long)blockIdx.x * blockDim.x + threadIdx.x;
  if (i >= n) return;
  outp[i] = in[i] + bias[i % F];
}

// ---------------- Layer-2 edge kernels (H=1, C=16) ----------------
__global__ void attn2_kernel(const float* __restrict__ xl, const float* __restrict__ xr,
                             const int* __restrict__ srcI, const int* __restrict__ dstI,
                             const float* __restrict__ att,
                             float* __restrict__ ebuf, float* __restrict__ mbuf,
                             long long Ereal, long long Etot) {
  long long e = (long long)blockIdx.x * blockDim.x + threadIdx.x;
  if (e >= Etot) return;
  int s, d;
  if (e < Ereal) { s = srcI[e]; d = dstI[e]; } else { s = d = (int)(e - Ereal); }
  const float* xs = xl + (size_t)s * 16;
  const float* xd = xr + (size_t)d * 16;
  float acc = 0.f;
#pragma unroll
  for (int f = 0; f < 16; ++f) {
    float v = xs[f] + xd[f];
    v = v > 0.f ? v : NEG_SLOPE * v;
    acc += v * att[f];
  }
  ebuf[e] = acc;
  atomicMaxF(&mbuf[d], acc);
}

__global__ void softden2_kernel(float* __restrict__ ebuf, float* __restrict__ den,
                                const int* __restrict__ dstI, const float* __restrict__ mbuf,
                                long long Ereal, long long Etot) {
  long long e = (long long)blockIdx.x * blockDim.x + threadIdx.x;
  if (e >= Etot) return;
  int d = (e < Ereal) ? dstI[e] : (int)(e - Ereal);
  float ex = expf(ebuf[e] - mbuf[d]);
  ebuf[e] = ex;
  atomicAdd(&den[d], ex);
}

__global__ void agg2_kernel(const float* __restrict__ xl, const float* __restrict__ ebuf,
                            const float* __restrict__ den,
                            const int* __restrict__ srcI, const int* __restrict__ dstI,
                            float* __restrict__ outp, long long Ereal, long long Etot) {
  long long idx = (long long)blockIdx.x * blockDim.x + threadIdx.x;
  long long e = idx >> 4;
  int f = (int)(idx & 15);
  if (e >= Etot) return;
  int s, d;
  if (e < Ereal) { s = srcI[e]; d = dstI[e]; } else { s = d = (int)(e - Ereal); }
  float alpha = ebuf[e] / (den[d] + 1e-16f);
  atomicAdd(&outp[(size_t)d * 16 + f], xl[(size_t)s * 16 + f] * alpha);
}

// ---------------- Final: log_softmax(y), log_softmax(z), 1 - cos(y,z) ----------------
__global__ void final_kernel(const float* __restrict__ y, const float* __restrict__ z,
                             float* __restrict__ outp, int N) {
  int n = blockIdx.x * blockDim.x + threadIdx.x;
  if (n >= N) return;
  float yv[16], zv[16];
#pragma unroll
  for (int i = 0; i < 16; ++i) { yv[i] = y[(size_t)n * 16 + i]; zv[i] = z[(size_t)n * 16 + i]; }
  float my = -INFINITY, mz = -INFINITY;
#pragma unroll
  for (int i = 0; i < 16; ++i) { my = fmaxf(my, yv[i]); mz = fmaxf(mz, zv[i]); }
  float sy = 0.f, sz = 0.f;
#pragma unroll
  for (int i = 0; i < 16; ++i) { sy += expf(yv[i] - my); sz += expf(zv[i] - mz); }
  float lsy = my + logf(sy), lsz = mz + logf(sz);
  float dot = 0.f, ny2 = 0.f, nz2 = 0.f;
#pragma unroll
  for (int i = 0; i < 16; ++i) { dot += yv[i] * zv[i]; ny2 += yv[i] * yv[i]; nz2 += zv[i] * zv[i]; }
  float ny = fmaxf(sqrtf(ny2), 1e-8f);
  float nz = fmaxf(sqrtf(nz2), 1e-8f);
  float cosv = dot / (ny * nz);

  size_t NN = (size_t)N;
  float* o0 = outp;            // logp_y
  float* o1 = outp + NN * 16;  // 1 - cos
  float* o2 = outp + NN * 17;  // logp_z
  float* o3 = outp + NN * 33;  // logp_y
  float* o4 = outp + NN * 49;  // logp_y
#pragma unroll
  for (int i = 0; i < 16; ++i) {
    float ly = yv[i] - lsy, lz = zv[i] - lsz;
    o0[(size_t)n * 16 + i] = ly;
    o2[(size_t)n * 16 + i] = lz;
    o3[(size_t)n * 16 + i] = ly;
    o4[(size_t)n * 16 + i] = ly;
  }
  o1[n] = 1.0f - cosv;
}

extern "C" void kernel_launch(void* const* d_in, const int* in_sizes, int n_in,
                              void* d_out, int out_size, void* d_ws, size_t ws_size,
                              hipStream_t stream) {
  const float* x1   = (const float*)d_in[0];
  const int*   ei1  = (const int*)d_in[1];
  const float* x2   = (const float*)d_in[2];
  const int*   ei2  = (const int*)d_in[3];
  const float* W1l  = (const float*)d_in[4];
  const float* W1r  = (const float*)d_in[5];
  const float* att1 = (const float*)d_in[6];
  const float* b1   = (const float*)d_in[7];
  const float* W2l  = (const float*)d_in[8];
  const float* W2r  = (const float*)d_in[9];
  const float* att2 = (const float*)d_in[10];
  const float* b2   = (const float*)d_in[11];

  const int Fin = 256, F1 = 256, C = 16;
  const long long N    = in_sizes[0] / Fin;
  const long long E    = in_sizes[1] / 2;
  const long long Etot = E + N;

  float* w = (float*)d_ws;
  float* xl   = w; w += (size_t)N * F1;
  float* xr   = w; w += (size_t)N * F1;
  float* agg  = w; w += (size_t)N * F1;   // becomes h after bias+elu
  float* ebuf = w; w += (size_t)Etot * 8;
  float* mbuf = w; w += (size_t)N * 8;
  float* den  = w; w += (size_t)N * 8;
  float* xl2  = w; w += (size_t)N * C;
  float* xr2  = w; w += (size_t)N * C;
  float* agg2 = w; w += (size_t)N * C;
  float* e2   = w; w += (size_t)Etot;
  float* m2   = w; w += (size_t)N;
  float* den2 = w; w += (size_t)N;
  float* ybuf = w; w += (size_t)N * C;
  float* zbuf = w; w += (size_t)N * C;

  auto cdiv = [](long long a, long long b) { return (int)((a + b - 1) / b); };

  const long long mTiles = (N + 31) / 32;   // 32-row wave tiles

  for (int g = 0; g < 2; ++g) {
    const float* x   = (g == 0) ? x1 : x2;
    const int* src   = (g == 0) ? ei1 : ei2;
    const int* dst   = src + E;
    float* yout      = (g == 0) ? ybuf : zbuf;

    // Layer 1 GEMMs (WMMA f32): xl = x@W1l, xr = x@W1r  (wave tile 32x64)
    {
      long long waves = mTiles * (F1 / 64);
      int blocks = cdiv(waves * 32, 256);
      gemm_wmma_f32<4><<<blocks, 256, 0, stream>>>(x, W1l, xl, (int)N, Fin, F1);
      gemm_wmma_f32<4><<<blocks, 256, 0, stream>>>(x, W1r, xr, (int)N, Fin, F1);
    }
    fill_kernel<<<cdiv(N * 8, 256), 256, 0, stream>>>(mbuf, N * 8, -INFINITY);
    fill_kernel<<<cdiv(N * 8, 256), 256, 0, stream>>>(den, N * 8, 0.f);
    fill_kernel<<<cdiv(N * F1, 256), 256, 0, stream>>>(agg, N * F1, 0.f);

    attn1_kernel<<<cdiv(Etot * 32, 256), 256, 0, stream>>>(xl, xr, src, dst, att1, ebuf, mbuf, E, Etot);
    softden1_kernel<<<cdiv(Etot, 256), 256, 0, stream>>>(ebuf, den, dst, mbuf, E, Etot);
    agg1_kernel<<<cdiv(Etot * 32, 256), 256, 0, stream>>>(xl, ebuf, den, src, dst, agg, E, Etot);
    bias_elu_kernel<<<cdiv(N * F1, 256), 256, 0, stream>>>(agg, b1, N * F1, F1);

    // Layer 2 GEMMs (WMMA f32): xl2 = h@W2l, xr2 = h@W2r  (wave tile 32x16)
    {
      int blocks = cdiv(mTiles * 32, 256);
      gemm_wmma_f32<1><<<blocks, 256, 0, stream>>>(agg, W2l, xl2, (int)N, F1, C);
      gemm_wmma_f32<1><<<blocks, 256, 0, stream>>>(agg, W2r, xr2, (int)N, F1, C);
    }
    fill_kernel<<<cdiv(N, 256), 256, 0, stream>>>(m2, N, -INFINITY);
    fill_kernel<<<cdiv(N, 256), 256, 0, stream>>>(den2, N, 0.f);
    fill_kernel<<<cdiv(N * C, 256), 256, 0, stream>>>(agg2, N * C, 0.f);

    attn2_kernel<<<cdiv(Etot, 256), 256, 0, stream>>>(xl2, xr2, src, dst, att2, e2, m2, E, Etot);
    softden2_kernel<<<cdiv(Etot, 256), 256, 0, stream>>>(e2, den2, dst, m2, E, Etot);
    agg2_kernel<<<cdiv(Etot * 16, 256), 256, 0, stream>>>(xl2, e2, den2, src, dst, agg2, E, Etot);
    bias_kernel<<<cdiv(N * C, 256), 256, 0, stream>>>(agg2, yout, b2, N * C, C);
  }

  final_kernel<<<cdiv(N, 256), 256, 0, stream>>>(ybuf, zbuf, (float*)d_out, (int)N);
}